// DistLoss_32762010533988
// MI455X (gfx1250) — compile-verified
//
#include <hip/hip_runtime.h>
#include <hip/hip_bf16.h>

// ---------------------------------------------------------------------------
// Problem constants (match reference)
// ---------------------------------------------------------------------------
#define NCL      32                    // NUM_CLUSTERS
#define DIM      32
#define NPTS     100000
#define NPAIR    496                   // C(32,2)
#define NTRIP    4960                  // C(32,3)
#define NCENT    5488                  // 32 + 496 + 4960 = 343 * 16
#define CHUNK_T  7                     // centroid tiles per LDS chunk
#define CHUNK_R  112                   // 7*16 centroids per chunk
#define NCHUNK   49                    // 343 / 7
#define NXT      6250                  // 100000 / 16 x-row tiles
#define XT_PER_WG 8                    // waves per workgroup
#define LDS_STRIDE 34                  // padded row stride (floats), 8B aligned
#define EPSF     1e-12f

typedef __attribute__((ext_vector_type(2))) float        v2f;
typedef __attribute__((ext_vector_type(8))) float        v8f;
typedef __attribute__((ext_vector_type(4))) unsigned int u32x4;
typedef __attribute__((ext_vector_type(8))) int          i32x8;
typedef __attribute__((ext_vector_type(4))) int          i32x4;

// Counter waits via inline asm (mnemonics verified in compiler-emitted disasm;
// the *builtin* forms are not declared on this toolchain).
#define WAIT_DSCNT0()    asm volatile("s_wait_dscnt 0" ::: "memory")
#define WAIT_ASYNCCNT0() asm volatile("s_wait_asynccnt 0" ::: "memory")

// ---------------------------------------------------------------------------
// Staging path selection: TDM > async global->LDS > synchronous fallback
// ---------------------------------------------------------------------------
#if __has_builtin(__builtin_amdgcn_tensor_load_to_lds) && \
    __has_builtin(__builtin_amdgcn_s_wait_tensorcnt)
#define USE_TDM 1
#else
#define USE_TDM 0
#endif

#if !USE_TDM && __has_builtin(__builtin_amdgcn_global_load_async_to_lds_b64)
#define USE_ASYNC_COPY 1
#else
#define USE_ASYNC_COPY 0
#endif

#if USE_TDM || USE_ASYNC_COPY
#define NBUF 2
#else
#define NBUF 1
#endif

#if USE_TDM
#pragma message("CDNA5 staging path: TDM tensor_load_to_lds (double-buffered)")
#elif USE_ASYNC_COPY
#pragma message("CDNA5 staging path: global_load_async_to_lds_b64 (double-buffered)")
#else
#pragma message("CDNA5 staging path: synchronous fallback + global_prefetch_b8")
#endif

#if USE_TDM
// One TDM descriptor copies a 112x32 f32 tile (row stride 32) into LDS with
// hardware padding: pad_interval=4 -> after every 32 DWORDs, pad_amount=1 ->
// +2 DWORDs, reproducing the stride-34 bank-conflict-free LDS layout.
__device__ __forceinline__ void tdm_load_chunk(const float* gsrc, float* ldst) {
    unsigned long long ga = (unsigned long long)gsrc;
    unsigned int       la = (unsigned int)(unsigned long long)ldst; // LDS byte off

    u32x4 g0;
    g0[0] = 1u;                                        // count=1, user, no gather
    g0[1] = la;                                        // lds_addr [63:32]
    g0[2] = (unsigned int)(ga & 0xFFFFFFFFu);          // global_addr[31:0]
    g0[3] = (unsigned int)((ga >> 32) & 0x01FFFFFFu)   // global_addr[56:32]
          | (2u << 30);                                // type=2 ("image")

    i32x8 g1;
    g1[0] = (2 << 16)                                  // data_size = 4 bytes
          | (1 << 20)                                  // pad_enable
          | (4 << 22)                                  // pad_interval: 32 DWORDs
          | (1 << 25);                                 // pad_amount: 2 DWORDs
    g1[1] = (DIM << 16);                               // tensor_dim0 = 32
    g1[2] = (NCENT << 16);                             // tensor_dim1 = 5488
    g1[3] = (DIM << 16);                               // tile_dim0 = 32
    g1[4] = CHUNK_R;                                   // tile_dim1 = 112, tile_dim2=0
    g1[5] = DIM;                                       // tensor_dim0_stride = 32
    g1[6] = 0;                                         // dim1_stride (unused, 2D)
    g1[7] = 0;

    i32x4 gz = {0, 0, 0, 0};                           // groups 2/3: dims unused
#if defined(__clang_major__) && __clang_major__ >= 23
    i32x8 gz8 = {0, 0, 0, 0, 0, 0, 0, 0};
    __builtin_amdgcn_tensor_load_to_lds(g0, g1, gz, gz, gz8, 0);
#else
    __builtin_amdgcn_tensor_load_to_lds(g0, g1, gz, gz, 0);
#endif
}
#endif

#if USE_ASYNC_COPY
__device__ __forceinline__ void async_copy_b64(const float* g, float* l) {
    const __attribute__((address_space(1))) void* gp =
        (const __attribute__((address_space(1))) void*)(unsigned long long)g;
    __attribute__((address_space(3))) void* lp =
        (__attribute__((address_space(3))) void*)(unsigned int)(unsigned long long)l;
    __builtin_amdgcn_global_load_async_to_lds_b64(gp, lp, 0, 0);
}

__device__ __forceinline__ void stage_chunk_async(const float* __restrict__ ncn,
                                                  const float* __restrict__ halfc,
                                                  int ch, float* lc, float* lh,
                                                  int tid) {
    const float* cg = ncn + (size_t)ch * CHUNK_R * DIM;
#pragma unroll
    for (int it = 0; it < (CHUNK_R * DIM / 2) / 256; ++it) {   // 7 b64/thread
        int f = (tid + it * 256) * 2;
        int r = f >> 5, c = f & 31;
        async_copy_b64(cg + f, lc + r * LDS_STRIDE + c);
    }
    if (tid < CHUNK_R) lh[tid] = halfc[ch * CHUNK_R + tid];
}
#endif

// ---------------------------------------------------------------------------
// g(X1, X2) row element:  X1@W1a.T + X2@W1a.T + (X1*X2)@W1b.T + 2*b1a + b1b
// ---------------------------------------------------------------------------
__device__ __forceinline__ float g_elem(const float* __restrict__ X1,
                                        const float* __restrict__ X2,
                                        const float* __restrict__ W1a,
                                        const float* __restrict__ b1a,
                                        const float* __restrict__ W1b,
                                        const float* __restrict__ b1b,
                                        int d) {
    float acc = 2.0f * b1a[d] + b1b[d];
    const float* wa = W1a + d * DIM;
    const float* wb = W1b + d * DIM;
#pragma unroll 8
    for (int k = 0; k < DIM; ++k) {
        float x1 = X1[k], x2 = X2[k];
        acc = fmaf(x1 + x2, wa[k], acc);
        acc = fmaf(x1 * x2, wb[k], acc);
    }
    return acc;
}

// ---------------------------------------------------------------------------
// Kernel 1: rows 0..31 copy W, rows 32..527 pair centroids. grid=528, block=32
// ---------------------------------------------------------------------------
__global__ void build_base_pairs(const float* __restrict__ W,
                                 const float* __restrict__ W1a,
                                 const float* __restrict__ b1a,
                                 const float* __restrict__ W1b,
                                 const float* __restrict__ b1b,
                                 float* __restrict__ C_all) {
    int r = blockIdx.x;
    int d = threadIdx.x;
    if (r < NCL) {
        C_all[r * DIM + d] = W[r * DIM + d];
        return;
    }
    int p = r - NCL;                   // pair index in triu order
    int i = 0;
    while (p >= (NCL - 1 - i)) { p -= (NCL - 1 - i); ++i; }
    int j = i + 1 + p;
    C_all[r * DIM + d] = g_elem(W + i * DIM, W + j * DIM, W1a, b1a, W1b, b1b, d);
}

// ---------------------------------------------------------------------------
// Kernel 2: rows 528..5487 triple centroids. grid=4960, block=32
// ---------------------------------------------------------------------------
__global__ void build_triples(const float* __restrict__ W,
                              const float* __restrict__ W1a,
                              const float* __restrict__ b1a,
                              const float* __restrict__ W1b,
                              const float* __restrict__ b1b,
                              float* __restrict__ C_all) {
    int t = blockIdx.x;                // triple index (combinations order)
    int d = threadIdx.x;
    int row = NCL + NPAIR + t;
    int i = 0;
    while (t >= ((NCL - 1 - i) * (NCL - 2 - i)) / 2) {
        t -= ((NCL - 1 - i) * (NCL - 2 - i)) / 2;
        ++i;
    }
    int j = i + 1;
    while (t >= (NCL - 1 - j)) { t -= (NCL - 1 - j); ++j; }
    int k = j + 1 + t;
    int p = i * (NCL - 1) - (i * (i - 1)) / 2 + (j - i - 1);   // pair idx of (i,j)
    const float* X1 = C_all + (NCL + p) * DIM;                 // pairs[p]
    const float* X2 = W + k * DIM;
    C_all[row * DIM + d] = g_elem(X1, X2, W1a, b1a, W1b, b1b, d);
}

// ---------------------------------------------------------------------------
// Kernel 3: per-centroid normalize. ncn = -c/max(|c|,eps); halfc = 0.5*|cn|^2
// ---------------------------------------------------------------------------
__global__ void finalize_centroids(const float* __restrict__ C_all,
                                   float* __restrict__ ncn,
                                   float* __restrict__ halfc) {
    int r = blockIdx.x;
    int d = threadIdx.x;
    float v = C_all[r * DIM + d];
    float s = v * v;
#pragma unroll
    for (int o = 16; o > 0; o >>= 1) s += __shfl_xor(s, o, 32);
    float norm = sqrtf(s);
    float inv = 1.0f / fmaxf(norm, EPSF);
    ncn[r * DIM + d] = -v * inv;                 // pre-negated for WMMA trick
    if (d == 0) halfc[r] = 0.5f * s * inv * inv; // 0.5*|cn|^2
}

__global__ void zero_accum(float* a) { *a = 0.0f; }

__global__ void final_sqrt(const float* a, float* out) { out[0] = sqrtf(fmaxf(*a, 0.0f)); }

// ---------------------------------------------------------------------------
// Per-chunk WMMA compute: 7 tiles x (8 chained V_WMMA_F32_16X16X4_F32) + argmin
// ---------------------------------------------------------------------------
__device__ __forceinline__ void compute_chunk(const float* __restrict__ lc,
                                              const float* __restrict__ lh,
                                              int mchunk_base,
                                              const v2f* __restrict__ bfrag,
                                              int half, int idx16,
                                              float& minval, int& minidx) {
    for (int tt = 0; tt < CHUNK_T; ++tt) {
        const int mbase = mchunk_base + tt * 16;
        v8f acc;                       // C preload: acc[r] <-> M = half*8 + r
#pragma unroll
        for (int r = 0; r < 8; ++r) acc[r] = lh[tt * 16 + half * 8 + r];

#pragma unroll
        for (int k0 = 0; k0 < 8; ++k0) {
            const int k = k0 * 4 + half * 2;
            v2f a;
            a.x = lc[(tt * 16 + idx16) * LDS_STRIDE + k];
            a.y = lc[(tt * 16 + idx16) * LDS_STRIDE + k + 1];
            acc = __builtin_amdgcn_wmma_f32_16x16x4_f32(
                false, a, false, bfrag[k0], (short)0, acc, false, false);
        }

#pragma unroll
        for (int r = 0; r < 8; ++r) {
            float v = acc[r];
            int   m = mbase + half * 8 + r;
            if (v < minval) { minval = v; minidx = m; }   // m ascending -> first-min
        }
    }
}

// ---------------------------------------------------------------------------
// Main kernel: per 16-point tile, sweep all 5488 centroids with fp32 WMMA.
//   score[m][n] = 0.5*|cn_m|^2 - cn_m . xn_n   (argmin == reference argmin)
// Double-buffered centroid chunks; TDM / async DMA to LDS when available.
// ---------------------------------------------------------------------------
__global__ void __launch_bounds__(256)
assign_kernel(const float* __restrict__ x,
              const float* __restrict__ ncn,
              const float* __restrict__ halfc,
              float* __restrict__ out_assign,   // d_out + 1, float-encoded index
              float* __restrict__ dist_accum) {
    __shared__ __align__(16) float lds_c[NBUF][CHUNK_R * LDS_STRIDE];
    __shared__ __align__(16) float lds_h[NBUF][CHUNK_R];
    __shared__ __align__(16) float lds_x[XT_PER_WG][16 * LDS_STRIDE];

    const int tid   = threadIdx.x;
    const int wave  = tid >> 5;
    const int lane  = tid & 31;
    const int half  = lane >> 4;        // 0: K/M low half, 1: high half
    const int idx16 = lane & 15;        // M for A-frag, N for B-frag

    const int  xtile  = blockIdx.x * XT_PER_WG + wave;
    const bool active = (xtile < NXT);
    const int  xt     = active ? xtile : (NXT - 1);        // clamp for loads

    // ---- stage this wave's 16x32 x block into LDS (coalesced) ----
    const float* xblk = x + (size_t)xt * 16 * DIM;
    float* xl = &lds_x[wave][0];
    for (int f = lane; f < 16 * DIM; f += 32) {
        int r = f >> 5, c = f & 31;
        xl[r * LDS_STRIDE + c] = xblk[f];
    }
    __syncthreads();

    // ---- row norm for n = idx16 (redundantly computed in both halves) ----
    float s = 0.0f;
#pragma unroll 8
    for (int c = 0; c < DIM; ++c) {
        float v = xl[idx16 * LDS_STRIDE + c];
        s += v * v;
    }
    float norm = sqrtf(s);
    float inv  = 1.0f / fmaxf(norm, EPSF);
    float xn2  = s * inv * inv;                            // |xn_n|^2

    // ---- 8 B fragments (xn tile, K=32) in registers ----
    v2f bfrag[8];
#pragma unroll
    for (int k0 = 0; k0 < 8; ++k0) {
        int k = k0 * 4 + half * 2;
        bfrag[k0].x = xl[idx16 * LDS_STRIDE + k]     * inv;
        bfrag[k0].y = xl[idx16 * LDS_STRIDE + k + 1] * inv;
    }

    float minval = 3.0e38f;
    int   minidx = 0;

#if USE_TDM
    // ---- pipelined: TDM DMA next chunk while computing current ----
    if (wave == 0) tdm_load_chunk(ncn, &lds_c[0][0]);
    if (tid < CHUNK_R) lds_h[0][tid] = halfc[tid];
    for (int ch = 0; ch < NCHUNK; ++ch) {
        const int cur = ch & 1;
        asm volatile("" ::: "memory");
        WAIT_DSCNT0();                         // our LDS reads/writes retired
        if (wave == 0)
            __builtin_amdgcn_s_wait_tensorcnt(0);  // chunk `cur` DMA complete
        __builtin_amdgcn_s_barrier();          // raw barrier: next DMA stays async
        asm volatile("" ::: "memory");
        if (ch + 1 < NCHUNK) {
            if (wave == 0)
                tdm_load_chunk(ncn + (size_t)(ch + 1) * CHUNK_R * DIM,
                               &lds_c[1 - cur][0]);
            if (tid < CHUNK_R)
                lds_h[1 - cur][tid] = halfc[(ch + 1) * CHUNK_R + tid];
        }
        compute_chunk(lds_c[cur], lds_h[cur], ch * CHUNK_R,
                      bfrag, half, idx16, minval, minidx);
    }
#elif USE_ASYNC_COPY
    // ---- pipelined: async-DMA next chunk while computing current ----
    stage_chunk_async(ncn, halfc, 0, lds_c[0], lds_h[0], tid);
    for (int ch = 0; ch < NCHUNK; ++ch) {
        const int cur = ch & 1;
        asm volatile("" ::: "memory");
        WAIT_ASYNCCNT0();                      // chunk `cur` DMA complete (this wave)
        WAIT_DSCNT0();                         // our LDS ops (reads + lh stores) done
        __builtin_amdgcn_s_barrier();          // raw barrier: keep next DMA in flight
        asm volatile("" ::: "memory");
        if (ch + 1 < NCHUNK)
            stage_chunk_async(ncn, halfc, ch + 1, lds_c[1 - cur], lds_h[1 - cur], tid);
        compute_chunk(lds_c[cur], lds_h[cur], ch * CHUNK_R,
                      bfrag, half, idx16, minval, minidx);
    }
#else
    // ---- fallback: synchronous staging + prefetch of next chunk ----
    for (int ch = 0; ch < NCHUNK; ++ch) {
        __syncthreads();
        const float* cg = ncn + (size_t)ch * CHUNK_R * DIM;
        for (int f = tid; f < CHUNK_R * DIM; f += 256) {
            int r = f >> 5, c = f & 31;
            lds_c[0][r * LDS_STRIDE + c] = cg[f];
            if (ch + 1 < NCHUNK)
                __builtin_prefetch(cg + CHUNK_R * DIM + f, 0, 3);  // global_prefetch_b8
        }
        if (tid < CHUNK_R) lds_h[0][tid] = halfc[ch * CHUNK_R + tid];
        __syncthreads();
        compute_chunk(lds_c[0], lds_h[0], ch * CHUNK_R,
                      bfrag, half, idx16, minval, minidx);
    }
#endif

    // ---- combine the two halves (each holds half the M values for its n) ---
    {
        float oval = __shfl_xor(minval, 16, 32);
        int   oidx = __shfl_xor(minidx, 16, 32);
        if (oval < minval || (oval == minval && oidx < minidx)) {
            minval = oval; minidx = oidx;
        }
    }

    // ---- outputs ----
    if (active && half == 0) {
        out_assign[(size_t)xt * 16 + idx16] = (float)minidx;
    }

    // per-point squared distance: |cn_m - xn_n|^2 = 2*score + |xn|^2
    float contrib = fmaxf(2.0f * minval + xn2, 0.0f);
    float c = (half == 0) ? contrib : 0.0f;   // halves hold duplicates
#pragma unroll
    for (int o = 16; o > 0; o >>= 1) c += __shfl_xor(c, o, 32);
    if (active && lane == 0) atomicAdd(dist_accum, c);
}

// ---------------------------------------------------------------------------
// Host-side launch
// ---------------------------------------------------------------------------
extern "C" void kernel_launch(void* const* d_in, const int* in_sizes, int n_in,
                              void* d_out, int out_size, void* d_ws, size_t ws_size,
                              hipStream_t stream) {
    (void)in_sizes; (void)n_in; (void)out_size; (void)ws_size;
    const float* x   = (const float*)d_in[0];
    const float* W   = (const float*)d_in[1];
    const float* W1a = (const float*)d_in[2];
    const float* b1a = (const float*)d_in[3];
    const float* W1b = (const float*)d_in[4];
    const float* b1b = (const float*)d_in[5];

    float* ws    = (float*)d_ws;
    float* C_all = ws;                         // NCENT*DIM
    float* ncn   = C_all + NCENT * DIM;        // NCENT*DIM
    float* halfc = ncn + NCENT * DIM;          // NCENT
    float* accum = halfc + NCENT;              // 1

    float* out = (float*)d_out;                // [0]=dists, [1..]=assignment

    build_base_pairs  <<<NCL + NPAIR, DIM, 0, stream>>>(W, W1a, b1a, W1b, b1b, C_all);
    build_triples     <<<NTRIP,       DIM, 0, stream>>>(W, W1a, b1a, W1b, b1b, C_all);
    finalize_centroids<<<NCENT,       DIM, 0, stream>>>(C_all, ncn, halfc);
    zero_accum        <<<1, 1, 0, stream>>>(accum);

    const int grid = (NXT + XT_PER_WG - 1) / XT_PER_WG;    // 782
    assign_kernel<<<grid, 256, 0, stream>>>(x, ncn, halfc, out + 1, accum);

    final_sqrt<<<1, 1, 0, stream>>>(accum, out);
}